// MyModel_16226386444980
// MI455X (gfx1250) — compile-verified
//
#include <hip/hip_runtime.h>
#include <math.h>

#define N_NODES 100000
#define N_EDGES 1600000
#define IN_CH   128
#define OUT_CH  64

typedef __attribute__((ext_vector_type(2))) float v2f;
typedef __attribute__((ext_vector_type(8))) float v8f;

// ---------------------------------------------------------------------------
// K0: zero the output accumulator and the degree buffer
// ---------------------------------------------------------------------------
__global__ void gcn_init_kernel(float* __restrict__ deg, float* __restrict__ out) {
    int i = blockIdx.x * blockDim.x + threadIdx.x;
    if (i < N_NODES * OUT_CH) out[i] = 0.0f;
    if (i < N_NODES)          deg[i] = 0.0f;
}

// ---------------------------------------------------------------------------
// K1: deg[col[e]] += edge_weight[e]   (f32 atomics, deg is 400KB -> L2)
// ---------------------------------------------------------------------------
__global__ void gcn_deg_kernel(const long long* __restrict__ ei,
                               const float* __restrict__ ew,
                               float* __restrict__ deg) {
    int e = blockIdx.x * blockDim.x + threadIdx.x;
    if (e < N_EDGES) {
        long long c = ei[(size_t)N_EDGES + e];  // col = edge_index[1][e]
        atomicAdd(&deg[c], ew[e]);
    }
}

// ---------------------------------------------------------------------------
// K2: in-place deg -> deg_inv_sqrt
// ---------------------------------------------------------------------------
__global__ void gcn_rsqrt_kernel(float* __restrict__ deg) {
    int i = blockIdx.x * blockDim.x + threadIdx.x;
    if (i < N_NODES) {
        float d = deg[i];
        deg[i] = (d > 0.0f) ? rsqrtf(d) : 0.0f;
    }
}

// ---------------------------------------------------------------------------
// K3: h = x @ W via V_WMMA_F32_16X16X4_F32 (full fp32, matches reference).
// Block = 128 threads = 4 waves; wave w computes the 16x16 tile at
// rows [16*blockIdx.x, +16), cols [16*w, +16); K=128 -> 32 WMMA ops.
// ---------------------------------------------------------------------------
__global__ __launch_bounds__(128) void gcn_gemm_kernel(
        const float* __restrict__ x,
        const float* __restrict__ W,
        float* __restrict__ h) {
    const int wave = threadIdx.x >> 5;
    const int lane = threadIdx.x & 31;
    const int hi   = lane >> 4;     // 0 or 1: which K-half of the fragment
    const int mr   = lane & 15;     // A: row-in-tile   B/D: col-in-tile
    const int nb   = wave * 16;     // N tile base (0,16,32,48)

    const size_t mbase = (size_t)blockIdx.x * 16;
    const float* xrow  = x + (mbase + (size_t)mr) * IN_CH;

    v8f c = {};
    #pragma unroll 4
    for (int k0 = 0; k0 < IN_CH; k0 += 4) {
        const int ka = k0 + 2 * hi;                 // even -> 8B aligned
        v2f a = *(const v2f*)(xrow + ka);           // x[mr][ka], x[mr][ka+1]
        v2f b;
        b.x = W[(size_t)ka       * OUT_CH + nb + mr];
        b.y = W[(size_t)(ka + 1) * OUT_CH + nb + mr];
        // (neg_a, A, neg_b, B, c_mod, C, reuse_a, reuse_b)
        c = __builtin_amdgcn_wmma_f32_16x16x4_f32(
                false, a, false, b, (short)0, c, false, false);
    }

    // D layout: VGPR r holds (row = 8*hi + r, col = mr) of the 16x16 tile
    #pragma unroll
    for (int r = 0; r < 8; ++r) {
        size_t mm = mbase + (size_t)(8 * hi + r);
        h[mm * OUT_CH + (size_t)(nb + mr)] = c[r];
    }
}

// ---------------------------------------------------------------------------
// K4: edge scatter. One wave per edge; lane l handles channels 2l, 2l+1.
// h-row gather is a B64 load that hits L2 (h = 25.6MB << 192MB L2);
// accumulation via global_atomic_add_f32 into L2-resident d_out.
// ---------------------------------------------------------------------------
__global__ __launch_bounds__(256) void gcn_scatter_kernel(
        const long long* __restrict__ ei,
        const float* __restrict__ ew,
        const float* __restrict__ dis,   // deg_inv_sqrt
        const float* __restrict__ h,
        float* __restrict__ out) {
    const int e    = blockIdx.x * (blockDim.x >> 5) + (threadIdx.x >> 5);
    const int lane = threadIdx.x & 31;
    if (e >= N_EDGES) return;

    const long long r = ei[e];                       // row = edge_index[0][e]
    const long long c = ei[(size_t)N_EDGES + e];     // col = edge_index[1][e]
    const float norm  = dis[r] * ew[e] * dis[c];     // wave-uniform broadcast loads

    v2f hv = *(const v2f*)(h + (size_t)r * OUT_CH + 2 * lane);
    float* op = out + (size_t)c * OUT_CH + 2 * lane;
    atomicAdd(op,     norm * hv.x);
    atomicAdd(op + 1, norm * hv.y);
}

// ---------------------------------------------------------------------------
// K5: out = sigmoid(out + b)
// ---------------------------------------------------------------------------
__global__ void gcn_final_kernel(const float* __restrict__ b,
                                 float* __restrict__ out) {
    int i = blockIdx.x * blockDim.x + threadIdx.x;
    if (i < N_NODES * OUT_CH) {
        float v = out[i] + b[i & (OUT_CH - 1)];
        out[i] = 1.0f / (1.0f + expf(-v));
    }
}

// ---------------------------------------------------------------------------
extern "C" void kernel_launch(void* const* d_in, const int* in_sizes, int n_in,
                              void* d_out, int out_size, void* d_ws, size_t ws_size,
                              hipStream_t stream) {
    const float*     x  = (const float*)d_in[0];      // [N, 128] f32
    const long long* ei = (const long long*)d_in[1];  // [2, E]  i64
    const float*     ew = (const float*)d_in[2];      // [E]     f32
    const float*     W  = (const float*)d_in[3];      // [128, 64] f32
    const float*     b  = (const float*)d_in[4];      // [64]    f32
    float*           out = (float*)d_out;             // [N, 64] f32

    // workspace: deg/deg_inv_sqrt (N f32), then h (N*64 f32), 256B aligned
    float* deg = (float*)d_ws;
    size_t deg_bytes = ((size_t)N_NODES * sizeof(float) + 255) & ~(size_t)255;
    float* h = (float*)((char*)d_ws + deg_bytes);

    // K0: init accumulators
    gcn_init_kernel<<<(N_NODES * OUT_CH + 255) / 256, 256, 0, stream>>>(deg, out);
    // K1: degree accumulation
    gcn_deg_kernel<<<(N_EDGES + 255) / 256, 256, 0, stream>>>(ei, ew, deg);
    // K2: deg -> rsqrt(deg) in place
    gcn_rsqrt_kernel<<<(N_NODES + 255) / 256, 256, 0, stream>>>(deg);
    // K3: h = x @ W  (fp32 WMMA; 6250 blocks x 4 waves)
    gcn_gemm_kernel<<<N_NODES / 16, 128, 0, stream>>>(x, W, h);
    // K4: edge gather-scale-scatter (1 wave per edge, 8 waves per block)
    gcn_scatter_kernel<<<(N_EDGES + 7) / 8, 256, 0, stream>>>(ei, ew, deg, h, out);
    // K5: bias + sigmoid
    gcn_final_kernel<<<(N_NODES * OUT_CH + 255) / 256, 256, 0, stream>>>(b, out);
}